// SourceBiasSeq_49469433315597
// MI455X (gfx1250) — compile-verified
//
#include <hip/hip_runtime.h>
#include <stdint.h>

#define DD 256
#define ROWP 272            // padded LDS row stride (floats): rows d, d+2 land on disjoint banks
#define CHUNK_ROWS 16
#define NCHUNKS (DD / CHUNK_ROWS)   // 16 chunks of 16 rows (16 KB each)

#if defined(__HIP_DEVICE_COMPILE__)
typedef __attribute__((ext_vector_type(2))) float v2f;
typedef __attribute__((ext_vector_type(8))) float v8f;

// Async DMA one 16-byte packet: global -> LDS, tracked by ASYNCcnt.
__device__ __forceinline__ void async_b128(uint32_t lds_byte_off, const void* gaddr) {
  asm volatile("global_load_async_to_lds_b128 %0, %1, off"
               :: "v"(lds_byte_off), "v"(gaddr) : "memory");
}
__device__ __forceinline__ void wait_async0() {
  asm volatile("s_wait_asynccnt 0x0" ::: "memory");
}
#endif

__global__ __launch_bounds__(256)
void SourceBiasSeq_kernel(const float* __restrict__ input,
                          const int*   __restrict__ urls,
                          const float* __restrict__ trans,
                          const float* __restrict__ bias,
                          float* __restrict__ out)
{
#if defined(__HIP_DEVICE_COMPILE__)
  __shared__ float lds_m[2][CHUNK_ROWS * ROWP];  // double-buffered matrix chunks
  __shared__ float lds_x[DD];                    // the token's x vector

  const int n = blockIdx.x;            // one workgroup per token
  const int t = threadIdx.x;           // 256 threads = 8 waves
  const int u = urls[n];
  const float* M = trans + (size_t)u * (DD * DD);

  // Stage x[n] into LDS (one float per thread).
  lds_x[t] = input[(size_t)n * DD + t];

  const uint32_t m0 = (uint32_t)(uintptr_t)&lds_m[0][0];
  const uint32_t m1 = (uint32_t)(uintptr_t)&lds_m[1][0];

  // Prologue: async-copy chunk 0 (16 rows x 256 f32 = 1024 float4s; 4 per thread).
#pragma unroll
  for (int i = 0; i < 4; ++i) {
    const int q = t + i * 256;
    const int row = q >> 6, c4 = q & 63;
    async_b128(m0 + (uint32_t)(row * ROWP + c4 * 4) * 4u,
               (const void*)(M + row * DD + c4 * 4));
  }

  const int wave = t >> 5;             // 0..7
  const int lane = t & 31;
  const int h    = lane >> 4;          // lane half: K-split per f32 WMMA layout
  const int lo   = lane & 15;
  const int e0a  = wave * 32;          // this wave's two 16-wide output-column tiles
  const int e0b  = wave * 32 + 16;

  v8f c0 = {};
  v8f c1 = {};

  for (int r = 0; r < NCHUNKS; ++r) {
    // Our async transfers for chunk r are done -> LDS writes visible; barrier makes
    // every wave's transfers + previous compute globally done.
    wait_async0();
    __syncthreads();

    // Kick off chunk r+1 into the buffer that everyone just finished reading (iter r-1).
    const int rn = r + 1;
    if (rn < NCHUNKS) {
      const uint32_t mb = (rn & 1) ? m1 : m0;
      const float* gsrc = M + rn * (CHUNK_ROWS * DD);
#pragma unroll
      for (int i = 0; i < 4; ++i) {
        const int q = t + i * 256;
        const int row = q >> 6, c4 = q & 63;
        async_b128(mb + (uint32_t)(row * ROWP + c4 * 4) * 4u,
                   (const void*)(gsrc + row * DD + c4 * 4));
      }
    }

    // Compute on chunk r: 4 K-steps of 4, two column tiles per wave.
    const float* buf = (r & 1) ? &lds_m[1][0] : &lds_m[0][0];
#pragma unroll
    for (int jp = 0; jp < 4; ++jp) {
      const int dl = 4 * jp + 2 * h;   // this lane-half's first row of the K=4 slab

      // A (16x4 f32, rows broadcast = x): lanes 0-15 carry K=0,1; lanes 16-31 carry K=2,3.
      const float2 av = *(const float2*)&lds_x[r * CHUNK_ROWS + dl];
      const v2f a = {av.x, av.y};

      // B (4x16 f32): VGPR0 = rows {0 | 2}, VGPR1 = rows {1 | 3} striped by lane half.
      const v2f b0 = { buf[dl * ROWP + e0a + lo],
                       buf[(dl + 1) * ROWP + e0a + lo] };
      c0 = __builtin_amdgcn_wmma_f32_16x16x4_f32(false, a, false, b0,
                                                 (short)0, c0, false, false);

      const v2f b1 = { buf[dl * ROWP + e0b + lo],
                       buf[(dl + 1) * ROWP + e0b + lo] };
      c1 = __builtin_amdgcn_wmma_f32_16x16x4_f32(false, a, false, b1,
                                                 (short)0, c1, false, false);
    }
  }

  // Every accumulator row holds the same y (A rows identical): VGPR0 of lanes 0-15
  // is y[e0+lo] for row 0, lanes 16-31 hold the row-8 copy. Lane half selects tile.
  const int   e   = (h ? e0b : e0a) + lo;
  const float acc = h ? c1[0] : c0[0];
  const float yb  = acc + bias[(size_t)u * DD + e];
  out[(size_t)n * DD + e] = tanhf(yb);
#endif // __HIP_DEVICE_COMPILE__
}

extern "C" void kernel_launch(void* const* d_in, const int* in_sizes, int n_in,
                              void* d_out, int out_size, void* d_ws, size_t ws_size,
                              hipStream_t stream) {
  (void)n_in; (void)out_size; (void)d_ws; (void)ws_size;
  const float* input = (const float*)d_in[0];
  const int*   urls  = (const int*)d_in[1];
  const float* trans = (const float*)d_in[2];
  const float* bias  = (const float*)d_in[3];
  float* out = (float*)d_out;

  const int N = in_sizes[1];           // B*S tokens (4096)
  SourceBiasSeq_kernel<<<N, 256, 0, stream>>>(input, urls, trans, bias, out);
}